// _FractalMaskGenerator_28312424415247
// MI455X (gfx1250) — compile-verified
//
#include <hip/hip_runtime.h>
#include <math.h>

typedef __attribute__((ext_vector_type(2))) float v2f;
typedef __attribute__((ext_vector_type(8))) float v8f;

#define D_FEAT 768
#define HW     64
#define NPIX   4096   // 64*64

// ---------------------------------------------------------------------------
// Kernel 1: energy[r] = mean over D of x[r,:]^2.
// One wave32 per row. Lane l loads float4 at element offset i*128 + l*4
// (contiguous 512B per wave per iteration -> fully coalesced b128 loads).
// Pure bandwidth phase: 402 MB read ~= 17us at 23.3 TB/s.
// ---------------------------------------------------------------------------
__global__ __launch_bounds__(256) void energy_kernel(const float* __restrict__ x,
                                                     float* __restrict__ energy,
                                                     int nrows) {
  int wave = (int)((blockIdx.x * blockDim.x + threadIdx.x) >> 5);
  int lane = threadIdx.x & 31;
  if (wave >= nrows) return;
  const float4* row = (const float4*)(x + (size_t)wave * D_FEAT);
  float s = 0.f;
#pragma unroll
  for (int i = 0; i < 6; ++i) {
    float4 v = row[i * 32 + lane];
    s = fmaf(v.x, v.x, s);
    s = fmaf(v.y, v.y, s);
    s = fmaf(v.z, v.z, s);
    s = fmaf(v.w, v.w, s);
  }
#pragma unroll
  for (int off = 16; off > 0; off >>= 1) s += __shfl_xor(s, off, 32);
  if (lane == 0) energy[wave] = s * (1.0f / (float)D_FEAT);
}

// ---------------------------------------------------------------------------
// Kernel 2: per batch (32 blocks, 256 threads = 8 waves).
// Box filter as P = W_k * E * W_k^T via V_WMMA_F32_16X16X4_F32.
// VGPR layouts per CDNA5 ISA 7.12.2:
//   A (16x4 f32):  lanes 0-15 -> M=lane, VGPR0/1 = K0/K1; lanes 16-31 -> K2/K3
//   B (4x16 f32):  VGPR0: lanes0-15 = row K0, lanes16-31 = row K2; VGPR1: K1/K3
//   C/D (16x16):   VGPR r: lanes0-15 -> M=r, lanes16-31 -> M=r+8, N=lane&15
// ---------------------------------------------------------------------------
__device__ __forceinline__ float sigmoid_f(float v) {
  return 1.0f / (1.0f + __expf(-v));
}

__global__ __launch_bounds__(256) void fractal_kernel(const float* __restrict__ energy,
                                                      float* __restrict__ out) {
  __shared__ float sE[NPIX];   // energy tile (stays live whole kernel)
  __shared__ float sW[NPIX];   // per-level averaging weights W_k
  __shared__ float sT[NPIX];   // T = W * E intermediate
  __shared__ float red[16];
  __shared__ float sMean, sInvStd;

  const int b    = blockIdx.x;
  const int tid  = threadIdx.x;
  const int lane = tid & 31;
  const int wv   = tid >> 5;
  const int half = lane >> 4;   // 0: lanes 0-15, 1: lanes 16-31
  const int l15  = lane & 15;

  // ---- stage energy in LDS ----
  for (int i = tid; i < NPIX; i += 256) sE[i] = energy[b * NPIX + i];
  __syncthreads();

  // ---- batch mean / unbiased std ----
  float s = 0.f, s2 = 0.f;
  for (int i = tid; i < NPIX; i += 256) {
    float v = sE[i];
    s += v;
    s2 = fmaf(v, v, s2);
  }
#pragma unroll
  for (int off = 16; off > 0; off >>= 1) {
    s  += __shfl_xor(s, off, 32);
    s2 += __shfl_xor(s2, off, 32);
  }
  if (lane == 0) { red[wv] = s; red[8 + wv] = s2; }
  __syncthreads();
  if (tid == 0) {
    float S = 0.f, S2 = 0.f;
    for (int i = 0; i < 8; ++i) { S += red[i]; S2 += red[8 + i]; }
    float mean = S / (float)NPIX;
    float var  = (S2 - S * S / (float)NPIX) / (float)(NPIX - 1);
    float sd   = fmaxf(sqrtf(fmaxf(var, 0.f)), 1e-6f);
    sMean = mean;
    sInvStd = 1.0f / sd;
  }
  __syncthreads();
  const float mean = sMean, invstd = sInvStd;

  // each wave owns two 16x16 output tiles of the 4x4 tile grid
  const int tiles[2] = { wv, wv + 8 };

  float acc[16];

  // ---- level 0 (k=1): pooled == E ----
#pragma unroll
  for (int t = 0; t < 2; ++t) {
    const int tt = tiles[t];
    const int tI = tt >> 2, tJ = tt & 3;
#pragma unroll
    for (int r = 0; r < 8; ++r) {
      const int row = tI * 16 + r + half * 8;
      const int col = tJ * 16 + l15;
      acc[t * 8 + r] = sigmoid_f((sE[row * HW + col] - mean) * invstd);
    }
  }

  // ---- levels 1..3 (k = 2,4,8): P = W * E * W^T via WMMA f32 ----
  for (int level = 1; level < 4; ++level) {
    const int k = 1 << level;
    const int p = k >> 1;

    __syncthreads();  // previous level's W/T fully consumed
    // build W_k[i][u] = 1/count_i on window [i-p, i-p+k) clipped to [0,64)
    for (int idx = tid; idx < NPIX; idx += 256) {
      const int i = idx >> 6, u = idx & 63;
      int lo = i - p;      if (lo < 0)  lo = 0;
      int hi = i - p + k;  if (hi > HW) hi = HW;
      sW[idx] = (u >= lo && u < hi) ? (1.0f / (float)(hi - lo)) : 0.0f;
    }
    __syncthreads();

    // pass 1: T = W * E
#pragma unroll
    for (int t = 0; t < 2; ++t) {
      const int tt = tiles[t];
      const int tI = tt >> 2, tJ = tt & 3;
      v8f c = {};
      for (int k0 = 0; k0 < HW; k0 += 4) {
        const int ka = k0 + half * 2;
        v2f a, bb;
        a.x  = sW[(tI * 16 + l15) * HW + ka];
        a.y  = sW[(tI * 16 + l15) * HW + ka + 1];
        bb.x = sE[ka * HW + tJ * 16 + l15];
        bb.y = sE[(ka + 1) * HW + tJ * 16 + l15];
        c = __builtin_amdgcn_wmma_f32_16x16x4_f32(false, a, false, bb,
                                                  (short)0, c, false, false);
      }
#pragma unroll
      for (int r = 0; r < 8; ++r) {
        const int row = tI * 16 + r + half * 8;
        const int col = tJ * 16 + l15;
        sT[row * HW + col] = c[r];
      }
    }
    __syncthreads();

    // pass 2: P = T * W^T ; sigmoid + accumulate in registers
#pragma unroll
    for (int t = 0; t < 2; ++t) {
      const int tt = tiles[t];
      const int tI = tt >> 2, tJ = tt & 3;
      v8f c = {};
      for (int k0 = 0; k0 < HW; k0 += 4) {
        const int ka = k0 + half * 2;
        v2f a, bb;
        a.x  = sT[(tI * 16 + l15) * HW + ka];
        a.y  = sT[(tI * 16 + l15) * HW + ka + 1];
        // B[k][n] = W^T[k][n] = W[n][k] : read W transposed from LDS
        bb.x = sW[(tJ * 16 + l15) * HW + ka];
        bb.y = sW[(tJ * 16 + l15) * HW + ka + 1];
        c = __builtin_amdgcn_wmma_f32_16x16x4_f32(false, a, false, bb,
                                                  (short)0, c, false, false);
      }
#pragma unroll
      for (int r = 0; r < 8; ++r) {
        acc[t * 8 + r] += sigmoid_f((c[r] - mean) * invstd);
      }
    }
  }

  // ---- average the 4 maps and write out ----
#pragma unroll
  for (int t = 0; t < 2; ++t) {
    const int tt = tiles[t];
    const int tI = tt >> 2, tJ = tt & 3;
#pragma unroll
    for (int r = 0; r < 8; ++r) {
      const int row = tI * 16 + r + half * 8;
      const int col = tJ * 16 + l15;
      out[b * NPIX + row * HW + col] = acc[t * 8 + r] * 0.25f;
    }
  }
}

extern "C" void kernel_launch(void* const* d_in, const int* in_sizes, int n_in,
                              void* d_out, int out_size, void* d_ws, size_t ws_size,
                              hipStream_t stream) {
  const float* features = (const float*)d_in[0];
  float* energy = (float*)d_ws;               // nrows floats of scratch

  const int nrows = in_sizes[0] / D_FEAT;     // B * N = 131072
  const int B     = nrows / NPIX;             // 32

  energy_kernel<<<(nrows + 7) / 8, 256, 0, stream>>>(features, energy, nrows);
  fractal_kernel<<<B, 256, 0, stream>>>(energy, (float*)d_out);
}